// SimpleGNN_51342039056528
// MI455X (gfx1250) — compile-verified
//
#include <hip/hip_runtime.h>
#include <hip/hip_bf16.h>
#include <math.h>

typedef float v2f __attribute__((ext_vector_type(2)));
typedef float v8f __attribute__((ext_vector_type(8)));

#define NN   100000
#define NE   1600000
#define NG   256
#define HID  128

static inline int cdiv_ll(long long a, int b) { return (int)((a + b - 1) / b); }

// ---------------------------------------------------------------- utilities
__global__ __launch_bounds__(256) void k_zero(float* p, long long n) {
  long long i = (long long)blockIdx.x * blockDim.x + threadIdx.x;
  if (i < n) p[i] = 0.f;
}

__global__ __launch_bounds__(256) void k_pad_x(const float* __restrict__ x, float* __restrict__ xp) {
  int i = blockIdx.x * blockDim.x + threadIdx.x;
  if (i >= NN * 8) return;
  int node = i >> 3, c = i & 7;
  xp[i] = (c < 6) ? x[node * 6 + c] : 0.f;
}

__global__ __launch_bounds__(256) void k_pad_w1(const float* __restrict__ w1, float* __restrict__ w1p) {
  int i = blockIdx.x * blockDim.x + threadIdx.x;
  if (i >= 8 * HID) return;
  int k = i >> 7;                      // w1 is row-major (6,128): flat index matches
  w1p[i] = (k < 6) ? w1[i] : 0.f;
}

// ---------------------------------------------------------------- degrees
__global__ __launch_bounds__(256) void k_deg(const int* __restrict__ ei, float* __restrict__ deg) {
  int e = blockIdx.x * blockDim.x + threadIdx.x;
  if (e >= NE) return;
  atomicAdd(&deg[ei[NE + e]], 1.f);    // dst row of edge_index
}

__global__ __launch_bounds__(256) void k_dinv(float* __restrict__ dinv) {
  int i = blockIdx.x * blockDim.x + threadIdx.x;
  if (i >= NN) return;
  dinv[i] = rsqrtf(dinv[i] + 1.f);     // +1 self-loop
}

// ---------------------------------------------------------------- WMMA GEMM
// C[M x 128] = A[M x K] @ W[K x 128]; K % 4 == 0, M % 16 == 0 (M = 100000 = 6250*16).
// Block = 256 threads = 8 waves; block owns 16 rows, wave w owns cols [16w,16w+16).
// f32 WMMA 16x16x4 fragment layout per ISA 7.12.2:
//   A (16x4): lane&15 = M row; VGPR0/1 hold K = kk, kk+1 where kk = (lane>>4)*2
//   B (4x16): lane&15 = N col; VGPR0/1 hold K = kk, kk+1
//   C (16x16): lane&15 = N col; VGPR r holds M = r + (lane>>4)*8
__global__ __launch_bounds__(256) void k_gemm_wmma(const float* __restrict__ A,
                                                   const float* __restrict__ W,
                                                   float* __restrict__ C,
                                                   int K) {
  const int wave = threadIdx.x >> 5;   // 0..7 -> column tile
  const int lane = threadIdx.x & 31;
  const int col  = lane & 15;
  const int hi   = lane >> 4;          // 0 or 1
  const int kk   = hi * 2;
  const int row0 = blockIdx.x * 16;
  const int n0   = wave * 16;
  const int arow = row0 + col;

  v8f acc = {};
  for (int k0 = 0; k0 < K; k0 += 4) {
    v2f a, b;
    const float* ap = A + (size_t)arow * K + (k0 + kk);
    a.x = ap[0];
    a.y = ap[1];
    b.x = W[(size_t)(k0 + kk)     * HID + n0 + col];
    b.y = W[(size_t)(k0 + kk + 1) * HID + n0 + col];
    acc = __builtin_amdgcn_wmma_f32_16x16x4_f32(
        /*neg_a=*/false, a, /*neg_b=*/false, b,
        /*c_mod=*/(short)0, acc, /*reuse_a=*/false, /*reuse_b=*/false);
  }

  float* cp = C + (size_t)(row0 + hi * 8) * HID + n0 + col;
#pragma unroll
  for (int r = 0; r < 8; ++r) cp[(size_t)r * HID] = acc[r];
}

// ---------------------------------------------------------------- aggregation F=128
__global__ __launch_bounds__(256) void k_selfloop(const float* __restrict__ h,
                                                  const float* __restrict__ dinv,
                                                  float* __restrict__ agg) {
  int i = blockIdx.x * blockDim.x + threadIdx.x;
  if (i >= NN * HID) return;
  float d = dinv[i >> 7];
  agg[i] = h[i] * d * d;
}

__global__ __launch_bounds__(256) void k_scatter128(const int* __restrict__ ei,
                                                    const float* __restrict__ h,
                                                    const float* __restrict__ dinv,
                                                    float* __restrict__ agg) {
  long long t = (long long)blockIdx.x * blockDim.x + threadIdx.x;
  if (t >= (long long)NE * 32) return;
  int e  = (int)(t >> 5);
  int c4 = ((int)t & 31) * 4;
  int s = ei[e], d = ei[NE + e];
  float norm = dinv[s] * dinv[d];
  const float4 v = *(const float4*)(h + (size_t)s * HID + c4);
  float* o = agg + (size_t)d * HID + c4;
  atomicAdd(o + 0, v.x * norm);
  atomicAdd(o + 1, v.y * norm);
  atomicAdd(o + 2, v.z * norm);
  atomicAdd(o + 3, v.w * norm);
}

__global__ __launch_bounds__(256) void k_bias_relu(float* __restrict__ h,
                                                   const float* __restrict__ b, int relu) {
  int i = blockIdx.x * blockDim.x + threadIdx.x;
  if (i >= NN * HID) return;
  float v = h[i] + b[i & (HID - 1)];
  h[i] = relu ? fmaxf(v, 0.f) : v;
}

// ---------------------------------------------------------------- layer 4 (Fout = 1)
__global__ __launch_bounds__(256) void k_dot_w4(const float* __restrict__ h,
                                                const float* __restrict__ w4,
                                                float* __restrict__ z) {
  int wid  = (int)(((long long)blockIdx.x * blockDim.x + threadIdx.x) >> 5);
  int lane = threadIdx.x & 31;
  if (wid >= NN) return;
  float4 a = ((const float4*)(h + (size_t)wid * HID))[lane];
  float4 b = ((const float4*)w4)[lane];
  float s = a.x * b.x + a.y * b.y + a.z * b.z + a.w * b.w;
#pragma unroll
  for (int off = 16; off > 0; off >>= 1) s += __shfl_down(s, off, 32);
  if (lane == 0) z[wid] = s;
}

__global__ __launch_bounds__(256) void k_self1(const float* __restrict__ z,
                                               const float* __restrict__ dinv,
                                               float* __restrict__ agg1) {
  int i = blockIdx.x * blockDim.x + threadIdx.x;
  if (i >= NN) return;
  float d = dinv[i];
  agg1[i] = z[i] * d * d;
}

__global__ __launch_bounds__(256) void k_scatter1(const int* __restrict__ ei,
                                                  const float* __restrict__ z,
                                                  const float* __restrict__ dinv,
                                                  float* __restrict__ agg1) {
  int e = blockIdx.x * blockDim.x + threadIdx.x;
  if (e >= NE) return;
  int s = ei[e], d = ei[NE + e];
  atomicAdd(&agg1[d], z[s] * dinv[s] * dinv[d]);
}

// ---------------------------------------------------------------- pooling + sigmoid
__global__ __launch_bounds__(256) void k_pool(const float* __restrict__ agg1,
                                              const float* __restrict__ b4,
                                              const int* __restrict__ batch,
                                              float* __restrict__ pooled,
                                              float* __restrict__ counts) {
  int i = blockIdx.x * blockDim.x + threadIdx.x;
  if (i >= NN) return;
  int g = batch[i];
  atomicAdd(&pooled[g], agg1[i] + b4[0]);
  atomicAdd(&counts[g], 1.f);
}

__global__ __launch_bounds__(256) void k_final(const float* __restrict__ pooled,
                                               const float* __restrict__ counts,
                                               float* __restrict__ out) {
  int g = blockIdx.x * blockDim.x + threadIdx.x;
  if (g >= NG) return;
  float v = pooled[g] / fmaxf(counts[g], 1.f);
  out[g] = 1.f / (1.f + expf(-v));
}

// ---------------------------------------------------------------- launcher
extern "C" void kernel_launch(void* const* d_in, const int* in_sizes, int n_in,
                              void* d_out, int out_size, void* d_ws, size_t ws_size,
                              hipStream_t stream) {
  const float* x    = (const float*)d_in[0];
  const int*   ei   = (const int*)d_in[1];
  const int*   batch= (const int*)d_in[2];
  const float* W1   = (const float*)d_in[3];
  const float* b1   = (const float*)d_in[4];
  const float* W2   = (const float*)d_in[5];
  const float* b2   = (const float*)d_in[6];
  const float* W3   = (const float*)d_in[7];
  const float* b3   = (const float*)d_in[8];
  const float* W4   = (const float*)d_in[9];
  const float* b4   = (const float*)d_in[10];
  float* out = (float*)d_out;

  // workspace layout (floats)
  float* W = (float*)d_ws;
  size_t off = 0;
  float* dinv = W + off; off += 100096;           // deg -> dinv in place
  float* xpad = W + off; off += (size_t)NN * 8;   // [N,8] padded input
  float* w1p  = W + off; off += 8 * HID;          // [8,128] padded W1
  float* big0 = W + off; off += (size_t)NN * HID;
  float* big1 = W + off; off += (size_t)NN * HID;
  float* big2 = W + off; off += (size_t)NN * HID;
  float* z    = W + off; off += 100096;
  float* agg1 = W + off; off += 100096;
  float* pooled = W + off; off += NG;
  float* counts = W + off; off += NG;

  const int T = 256;
  const int gN    = cdiv_ll(NN, T);
  const int gE    = cdiv_ll(NE, T);
  const int gNH   = cdiv_ll((long long)NN * HID, T);
  const int gE32  = cdiv_ll((long long)NE * 32, T);
  const int gGemm = NN / 16;                      // 6250, exact

  // degrees
  k_zero<<<gN, T, 0, stream>>>(dinv, NN);
  k_deg<<<gE, T, 0, stream>>>(ei, dinv);
  k_dinv<<<gN, T, 0, stream>>>(dinv);

  // pad inputs for the K=8 WMMA GEMM
  k_pad_x<<<cdiv_ll((long long)NN * 8, T), T, 0, stream>>>(x, xpad);
  k_pad_w1<<<cdiv_ll(8 * HID, T), T, 0, stream>>>(W1, w1p);

  // ---- layer 1: h1 = relu(agg(xpad @ W1p) + b1) -> big1
  k_gemm_wmma<<<gGemm, T, 0, stream>>>(xpad, w1p, big0, 8);
  k_selfloop<<<gNH, T, 0, stream>>>(big0, dinv, big1);
  k_scatter128<<<gE32, T, 0, stream>>>(ei, big0, dinv, big1);
  k_bias_relu<<<gNH, T, 0, stream>>>(big1, b1, 1);

  // ---- layer 2: big1 @ W2 -> big0; agg -> big2
  k_gemm_wmma<<<gGemm, T, 0, stream>>>(big1, W2, big0, HID);
  k_selfloop<<<gNH, T, 0, stream>>>(big0, dinv, big2);
  k_scatter128<<<gE32, T, 0, stream>>>(ei, big0, dinv, big2);
  k_bias_relu<<<gNH, T, 0, stream>>>(big2, b2, 1);

  // ---- layer 3: big2 @ W3 -> big0; agg -> big1
  k_gemm_wmma<<<gGemm, T, 0, stream>>>(big2, W3, big0, HID);
  k_selfloop<<<gNH, T, 0, stream>>>(big0, dinv, big1);
  k_scatter128<<<gE32, T, 0, stream>>>(ei, big0, dinv, big1);
  k_bias_relu<<<gNH, T, 0, stream>>>(big1, b3, 1);

  // ---- layer 4: z = big1 . W4; scalar aggregate; pool; sigmoid
  k_dot_w4<<<cdiv_ll((long long)NN * 32, T), T, 0, stream>>>(big1, W4, z);
  k_self1<<<gN, T, 0, stream>>>(z, dinv, agg1);
  k_scatter1<<<gE, T, 0, stream>>>(ei, z, dinv, agg1);

  k_zero<<<cdiv_ll(2 * NG, T), T, 0, stream>>>(pooled, 2 * NG);  // pooled+counts contiguous
  k_pool<<<gN, T, 0, stream>>>(agg1, b4, batch, pooled, counts);
  k_final<<<cdiv_ll(NG, T), T, 0, stream>>>(pooled, counts, out);
}